// CIFSampler_23948737642795
// MI455X (gfx1250) — compile-verified
//
#include <hip/hip_runtime.h>
#include <cstdint>
#include <cstddef>

#define CIF_T 4096
#define CIF_B 16
#define CIF_D 512

// 128-bit payload type the async builtins expect (per hipcc diagnostic:
// "__attribute__((__vector_size__(4 * sizeof(int)))) int *").
typedef int cif_v4i __attribute__((vector_size(16)));

// ---------------------------------------------------------------------------
// CDNA5 async global<->LDS path (ASYNCcnt-tracked data mover), guarded so the
// host pass / older toolchains fall back to plain 128-bit copies.
// ---------------------------------------------------------------------------
#if defined(__gfx1250__) &&                                                   \
    __has_builtin(__builtin_amdgcn_global_load_async_to_lds_b128) &&          \
    __has_builtin(__builtin_amdgcn_global_store_async_from_lds_b128) &&       \
    __has_builtin(__builtin_amdgcn_s_wait_asynccnt)
#define CIF_USE_ASYNC_LDS 1
#else
#define CIF_USE_ASYNC_LDS 0
#endif

// ---------------------------------------------------------------------------
// Kernel 1: integrate-and-fire scan + index compaction.
// Single wave32 block. alphas is [B][T]; we tile 32 time-steps at a time:
// coalesced global loads (lanes sweep t), LDS tile stored transposed [tt][b]
// so the scan read (lane == b) is bank-conflict free. Lanes 16..31 integrate
// a = 0 and never fire, so the full-wave __any() is exact.
// ---------------------------------------------------------------------------
__global__ __launch_bounds__(32) void cif_scan_kernel(
    const float* __restrict__ alphas,  // [B][T]
    int* __restrict__ idx,             // [T] compacted fired time indices
    float* __restrict__ num_out)       // scalar (written as float)
{
    __shared__ float tile[32 * CIF_B];  // [tt][b]
    const int lane = threadIdx.x;       // 0..31 (wave32)

    float integrated = 0.0f;
    int cnt = 0;

    for (int t0 = 0; t0 < CIF_T; t0 += 32) {
        // Cooperative coalesced load of a 16 x 32 alpha tile, transposed.
#pragma unroll
        for (int b = 0; b < CIF_B; ++b) {
            tile[lane * CIF_B + b] = alphas[b * CIF_T + t0 + lane];
        }
        __syncthreads();

#pragma unroll 4
        for (int tt = 0; tt < 32; ++tt) {
            const float a = (lane < CIF_B) ? tile[tt * CIF_B + lane] : 0.0f;
            integrated += a;
            const bool fire = (integrated >= 1.0f);
            if (fire) integrated -= 1.0f;           // == integrated - fire*THRESHOLD
            if (__any((int)fire)) {                 // OR across batch lanes
                if (lane == 0) idx[cnt] = t0 + tt;
                ++cnt;                              // uniform across wave
            }
        }
        __syncthreads();
    }

    // Pad remainder of idx with 0 (reference nonzero fill_value=0).
    for (int j = cnt + lane; j < CIF_T; j += 32) idx[j] = 0;
    if (lane == 0) *num_out = (float)cnt;
}

// ---------------------------------------------------------------------------
// Kernel 2: row gather  out[b, j, :] = enc[b, idx[j], :].
// One block per 2KB row; 128 lanes x 16B. On gfx1250 the row is streamed
// through LDS with the async data mover (no VGPR round trip):
//   global_load_async_to_lds_b128 -> s_wait_asynccnt 0 -> barrier ->
//   global_store_async_from_lds_b128  (s_endpgm drains ASYNCcnt).
// ---------------------------------------------------------------------------
__global__ __launch_bounds__(128) void cif_gather_kernel(
    const float* __restrict__ enc,   // [B][T][D]
    const int* __restrict__ idx,     // [T]
    float* __restrict__ out)         // [B][T][D]
{
    const int j = blockIdx.x;        // output time slot
    const int b = blockIdx.y;        // batch
    const int tid = threadIdx.x;     // 0..127, 16 bytes each

    const int src_t = idx[j];        // uniform scalar load
    const float* __restrict__ src = enc + ((size_t)b * CIF_T + src_t) * CIF_D;
    float* __restrict__ dst       = out + ((size_t)b * CIF_T + j) * CIF_D;

#if CIF_USE_ASYNC_LDS
    __shared__ float buf[CIF_D];     // 2 KB staging tile
    __builtin_amdgcn_global_load_async_to_lds_b128(
        (__attribute__((address_space(1))) cif_v4i*)(src + 4 * tid),
        (__attribute__((address_space(3))) cif_v4i*)(&buf[4 * tid]),
        /*imm offset*/ 0, /*cpol*/ 0);
    __builtin_amdgcn_s_wait_asynccnt(0);   // this wave's loads landed in LDS
    __syncthreads();                       // all waves' loads landed
    __builtin_amdgcn_global_store_async_from_lds_b128(
        (__attribute__((address_space(1))) cif_v4i*)(dst + 4 * tid),
        (__attribute__((address_space(3))) cif_v4i*)(&buf[4 * tid]),
        /*imm offset*/ 0, /*cpol*/ 0);
    // s_endpgm performs an implicit wait-idle, draining ASYNCcnt.
#else
    const float4 v = ((const float4*)src)[tid];
    ((float4*)dst)[tid] = v;
#endif
}

// ---------------------------------------------------------------------------
// Launch: inputs are (encoder_output f32 [16,4096,512], alphas f32 [16,4096]).
// d_out = 16*4096*512 sampled features + 1 num_sampled (float), flat.
// d_ws  = idx[4096] (16 KB).
// ---------------------------------------------------------------------------
extern "C" void kernel_launch(void* const* d_in, const int* in_sizes, int n_in,
                              void* d_out, int out_size, void* d_ws, size_t ws_size,
                              hipStream_t stream) {
    (void)in_sizes; (void)n_in; (void)ws_size;
    const float* enc    = (const float*)d_in[0];
    const float* alphas = (const float*)d_in[1];
    float* out          = (float*)d_out;
    int*   idx          = (int*)d_ws;

    // num_sampled is the last flat element of the concatenated outputs.
    float* num_out = out + ((size_t)out_size - 1);

    cif_scan_kernel<<<1, 32, 0, stream>>>(alphas, idx, num_out);

    dim3 grid(CIF_T, CIF_B, 1);
    cif_gather_kernel<<<grid, 128, 0, stream>>>(enc, idx, out);
}